// SelectiveSSM_19112604467502
// MI455X (gfx1250) — compile-verified
//
#include <hip/hip_runtime.h>

// Problem dims (match reference)
constexpr int BATCH  = 2;
constexpr int SEQ    = 2048;
constexpr int DMODEL = 1024;
constexpr int DSTATE = 16;

// Chunked scan config: 32 chunks of 64 steps -> 65536 independent threads in pass1/3
constexpr int NCHUNK = 32;
constexpr int TCH    = SEQ / NCHUNK;   // 64
constexpr int BLOCK_D = 256;           // threads per block (8 waves of 32)

#define F_LOG2E 1.44269504088896340736f

__device__ __forceinline__ float fexp2(float v) {
#if __has_builtin(__builtin_amdgcn_exp2f)
  return __builtin_amdgcn_exp2f(v);    // v_exp_f32
#else
  return exp2f(v);
#endif
}

#if __has_builtin(__builtin_amdgcn_global_load_async_to_lds_b128) && \
    __has_builtin(__builtin_amdgcn_s_wait_asynccnt)
#define USE_ASYNC_LDS 1
#else
#define USE_ASYNC_LDS 0
#endif

// Raw vector type + address-space helpers for the async builtin.
typedef int v4i_cdna __attribute__((ext_vector_type(4)));
#define GAS1 __attribute__((address_space(1)))
#define LAS3 __attribute__((address_space(3)))

// ---------------------------------------------------------------------------
// Pass 1: per-chunk local recurrence with zero init.
// Stores local final state (16 floats) and sum(dt) per (b, chunk, d).
// Transition over a chunk is exp(A[n] * sum(dt)) -- only sum(dt) is needed.
// B tile (64x16 f32 = 4KB) staged to LDS with async-to-LDS, overlapped with
// the A-row load + v_exp_f32 setup.
// ---------------------------------------------------------------------------
__global__ __launch_bounds__(BLOCK_D) void ssm_pass1(
    const float* __restrict__ x, const float* __restrict__ dt,
    const float* __restrict__ Bm, const float* __restrict__ Alog,
    float* __restrict__ sloc, float* __restrict__ sdt)
{
  __shared__ float4 sB[TCH * DSTATE / 4];   // 4 KB

  const int tid = threadIdx.x;
  const int d = blockIdx.x * BLOCK_D + tid;
  const int b = blockIdx.y;
  const int c = blockIdx.z;

  const size_t bc = ((size_t)b * SEQ + (size_t)c * TCH) * DSTATE;  // B tile base

  // Issue async tile copy first (256 threads x 16B = 4KB)...
#if USE_ASYNC_LDS
  __builtin_amdgcn_global_load_async_to_lds_b128(
      (GAS1 v4i_cdna*)(Bm + bc) + tid,
      (LAS3 v4i_cdna*)sB + tid, 0, 0);
#else
  sB[tid] = ((const float4*)(Bm + bc))[tid];
#endif

  // ...and overlap it with the A-row load + exp2 setup.
  // A (log2-scaled): An2[n] = -exp(A_log[d,n]) * log2(e)
  float An2[DSTATE];
  const float4* A4 = (const float4*)(Alog + (size_t)d * DSTATE);
#pragma unroll
  for (int i = 0; i < 4; ++i) {
    float4 v = A4[i];
    An2[4*i+0] = -fexp2(v.x * F_LOG2E) * F_LOG2E;
    An2[4*i+1] = -fexp2(v.y * F_LOG2E) * F_LOG2E;
    An2[4*i+2] = -fexp2(v.z * F_LOG2E) * F_LOG2E;
    An2[4*i+3] = -fexp2(v.w * F_LOG2E) * F_LOG2E;
  }

  float st[DSTATE];
#pragma unroll
  for (int n = 0; n < DSTATE; ++n) st[n] = 0.f;
  float sacc = 0.f;

#if USE_ASYNC_LDS
  __builtin_amdgcn_s_wait_asynccnt(0);
#endif
  __syncthreads();

  const size_t base = ((size_t)b * SEQ + (size_t)c * TCH) * DMODEL + d;
  const float* dtp = dt + base;
  const float* xp  = x  + base;

  for (int t = 0; t < TCH; ++t) {
    __builtin_prefetch(dtp + (size_t)(t + 8) * DMODEL, 0, 1);  // global_prefetch_b8
    __builtin_prefetch(xp  + (size_t)(t + 8) * DMODEL, 0, 1);
    const float dtv = dtp[(size_t)t * DMODEL];
    const float xv  = xp [(size_t)t * DMODEL];
    float bb[DSTATE];
#pragma unroll
    for (int i = 0; i < 4; ++i) {
      float4 v = sB[t * 4 + i];   // ds_load_b128, broadcast (conflict-free)
      bb[4*i+0] = v.x; bb[4*i+1] = v.y; bb[4*i+2] = v.z; bb[4*i+3] = v.w;
    }
    const float dtx = dtv * xv;
    sacc += dtv;
#pragma unroll
    for (int n = 0; n < DSTATE; ++n)
      st[n] = fexp2(dtv * An2[n]) * st[n] + dtx * bb[n];
  }

  float4* sl = (float4*)(sloc + (((size_t)b * NCHUNK + c) * DMODEL + d) * DSTATE);
#pragma unroll
  for (int i = 0; i < 4; ++i) {
    float4 v; v.x = st[4*i+0]; v.y = st[4*i+1]; v.z = st[4*i+2]; v.w = st[4*i+3];
    sl[i] = v;
  }
  sdt[((size_t)b * NCHUNK + c) * DMODEL + d] = sacc;
}

// ---------------------------------------------------------------------------
// Pass 2: tiny scan over chunks. One thread per (b, d, n) = 32768 threads.
// init[b,c,d,n] = state entering chunk c.
// ---------------------------------------------------------------------------
__global__ __launch_bounds__(256) void ssm_pass2(
    const float* __restrict__ Alog, const float* __restrict__ sloc,
    const float* __restrict__ sdt, float* __restrict__ init)
{
  const int tid = blockIdx.x * 256 + threadIdx.x;
  const int n = tid & (DSTATE - 1);
  const int d = (tid / DSTATE) & (DMODEL - 1);
  const int b = tid / (DSTATE * DMODEL);

  const float An2 = -fexp2(Alog[(size_t)d * DSTATE + n] * F_LOG2E) * F_LOG2E;
  float s0 = 0.f;
  for (int c = 0; c < NCHUNK; ++c) {
    const size_t idx = ((size_t)b * NCHUNK + c) * DMODEL + d;
    init[idx * DSTATE + n] = s0;
    s0 = fexp2(An2 * sdt[idx]) * s0 + sloc[idx * DSTATE + n];
  }
}

// ---------------------------------------------------------------------------
// Pass 3: re-run local recurrence with correct init, emit y = C·s + x*D.
// B and C chunk tiles (4 KB each) staged into LDS via async-to-LDS loads,
// overlapped with init-state + A-row loads and exp2 setup.
// ---------------------------------------------------------------------------
__global__ __launch_bounds__(BLOCK_D) void ssm_pass3(
    const float* __restrict__ x, const float* __restrict__ dt,
    const float* __restrict__ Bm, const float* __restrict__ Cm,
    const float* __restrict__ Alog, const float* __restrict__ Dv,
    const float* __restrict__ init, float* __restrict__ out)
{
  __shared__ float4 sB[TCH * DSTATE / 4];   // 4 KB
  __shared__ float4 sC[TCH * DSTATE / 4];   // 4 KB

  const int tid = threadIdx.x;
  const int d = blockIdx.x * BLOCK_D + tid;
  const int b = blockIdx.y;
  const int c = blockIdx.z;

  const size_t bc = ((size_t)b * SEQ + (size_t)c * TCH) * DSTATE;  // B/C tile base

  // Issue both async tile copies first (256 threads x 16B covers each 4KB tile)...
#if USE_ASYNC_LDS
  __builtin_amdgcn_global_load_async_to_lds_b128(
      (GAS1 v4i_cdna*)(Bm + bc) + tid,
      (LAS3 v4i_cdna*)sB + tid, 0, 0);
  __builtin_amdgcn_global_load_async_to_lds_b128(
      (GAS1 v4i_cdna*)(Cm + bc) + tid,
      (LAS3 v4i_cdna*)sC + tid, 0, 0);
#else
  sB[tid] = ((const float4*)(Bm + bc))[tid];
  sC[tid] = ((const float4*)(Cm + bc))[tid];
#endif

  // ...overlap with init-state load, D load, A-row load + exp2 setup.
  float An2[DSTATE];
  const float4* A4 = (const float4*)(Alog + (size_t)d * DSTATE);
#pragma unroll
  for (int i = 0; i < 4; ++i) {
    float4 v = A4[i];
    An2[4*i+0] = -fexp2(v.x * F_LOG2E) * F_LOG2E;
    An2[4*i+1] = -fexp2(v.y * F_LOG2E) * F_LOG2E;
    An2[4*i+2] = -fexp2(v.z * F_LOG2E) * F_LOG2E;
    An2[4*i+3] = -fexp2(v.w * F_LOG2E) * F_LOG2E;
  }

  float st[DSTATE];
  const float4* ip = (const float4*)(init + (((size_t)b * NCHUNK + c) * DMODEL + d) * DSTATE);
#pragma unroll
  for (int i = 0; i < 4; ++i) {
    float4 v = ip[i];
    st[4*i+0] = v.x; st[4*i+1] = v.y; st[4*i+2] = v.z; st[4*i+3] = v.w;
  }
  const float Dd = Dv[d];

#if USE_ASYNC_LDS
  __builtin_amdgcn_s_wait_asynccnt(0);
#endif
  __syncthreads();

  const size_t base = ((size_t)b * SEQ + (size_t)c * TCH) * DMODEL + d;
  const float* dtp = dt + base;
  const float* xp  = x  + base;
  float* outp      = out + base;

  for (int t = 0; t < TCH; ++t) {
    __builtin_prefetch(dtp + (size_t)(t + 8) * DMODEL, 0, 1);
    __builtin_prefetch(xp  + (size_t)(t + 8) * DMODEL, 0, 1);
    const float dtv = dtp[(size_t)t * DMODEL];
    const float xv  = xp [(size_t)t * DMODEL];
    float bb[DSTATE], cc[DSTATE];
#pragma unroll
    for (int i = 0; i < 4; ++i) {
      float4 vb = sB[t * 4 + i];   // ds_load_b128, broadcast (conflict-free)
      float4 vc = sC[t * 4 + i];
      bb[4*i+0] = vb.x; bb[4*i+1] = vb.y; bb[4*i+2] = vb.z; bb[4*i+3] = vb.w;
      cc[4*i+0] = vc.x; cc[4*i+1] = vc.y; cc[4*i+2] = vc.z; cc[4*i+3] = vc.w;
    }
    const float dtx = dtv * xv;
    float y = 0.f;
#pragma unroll
    for (int n = 0; n < DSTATE; ++n) {
      st[n] = fexp2(dtv * An2[n]) * st[n] + dtx * bb[n];
      y = fmaf(st[n], cc[n], y);
    }
    outp[(size_t)t * DMODEL] = fmaf(xv, Dd, y);
  }
}

// ---------------------------------------------------------------------------
extern "C" void kernel_launch(void* const* d_in, const int* in_sizes, int n_in,
                              void* d_out, int out_size, void* d_ws, size_t ws_size,
                              hipStream_t stream) {
  (void)in_sizes; (void)n_in; (void)out_size; (void)ws_size;
  const float* x    = (const float*)d_in[0];
  const float* dt   = (const float*)d_in[1];
  const float* Bm   = (const float*)d_in[2];
  const float* Cm   = (const float*)d_in[3];
  const float* Alog = (const float*)d_in[4];
  const float* Dv   = (const float*)d_in[5];
  float* out = (float*)d_out;

  float* ws   = (float*)d_ws;
  float* sloc = ws;                                                  // B*NC*DM*DN = 4 MB
  float* sdt  = sloc + (size_t)BATCH * NCHUNK * DMODEL * DSTATE;     // B*NC*DM    = 256 KB
  float* init = sdt  + (size_t)BATCH * NCHUNK * DMODEL;              // B*NC*DM*DN = 4 MB

  dim3 grid(DMODEL / BLOCK_D, BATCH, NCHUNK);
  ssm_pass1<<<grid, BLOCK_D, 0, stream>>>(x, dt, Bm, Alog, sloc, sdt);

  const int total2 = BATCH * DMODEL * DSTATE;   // 32768
  ssm_pass2<<<total2 / 256, 256, 0, stream>>>(Alog, sloc, sdt, init);

  ssm_pass3<<<grid, BLOCK_D, 0, stream>>>(x, dt, Bm, Cm, Alog, Dv, init, out);
}